// STransformer_40432822124869
// MI455X (gfx1250) — compile-verified
//
#include <hip/hip_runtime.h>
#include <math.h>

// ---- problem constants (match reference) ----
#define NN_   512   // nodes
#define TT_   64    // timesteps
#define CC_   64    // channels
#define HH_   4     // heads
#define HD_   16    // head dim
#define FH_   256   // FFN hidden (FE*C)
#define NT_   (NN_*TT_)              // 32768 rows
#define SLOT  ((size_t)NN_*TT_*CC_)  // 2097152 floats per buffer slot

typedef float v2f __attribute__((ext_vector_type(2)));
typedef float v8f __attribute__((ext_vector_type(8)));

static __device__ __forceinline__ v8f wmma4(v2f a, v2f b, v8f c) {
    // V_WMMA_F32_16X16X4_F32 : D = A(16x4,f32) x B(4x16,f32) + C(16x16,f32)
    return __builtin_amdgcn_wmma_f32_16x16x4_f32(
        false, a, false, b, (short)0, c, false, false);
}

// ---------------------------------------------------------------------------
// adj InstanceNorm statistics (mean, rstd over whole 512x512 map)
// ---------------------------------------------------------------------------
__global__ __launch_bounds__(256) void adj_stats_kernel(const float* __restrict__ adj,
                                                        float* __restrict__ stats) {
    __shared__ float r1[256];
    __shared__ float r2[256];
    const int n = NN_ * NN_;
    float s1 = 0.f, s2 = 0.f;
    for (int i = threadIdx.x; i < n; i += 256) {
        float a = adj[i];
        s1 += a; s2 += a * a;
    }
    r1[threadIdx.x] = s1; r2[threadIdx.x] = s2;
    __syncthreads();
    for (int off = 128; off > 0; off >>= 1) {
        if (threadIdx.x < off) {
            r1[threadIdx.x] += r1[threadIdx.x + off];
            r2[threadIdx.x] += r2[threadIdx.x + off];
        }
        __syncthreads();
    }
    if (threadIdx.x == 0) {
        float mean = r1[0] / (float)n;
        float var  = r2[0] / (float)n - mean * mean;
        stats[0] = mean;
        stats[1] = rsqrtf(var + 1e-5f);
    }
}

__global__ __launch_bounds__(256) void adj_norm_kernel(const float* __restrict__ adj,
                                                       const float* __restrict__ stats,
                                                       float* __restrict__ adjn) {
    const float mean = stats[0], rstd = stats[1];
    for (int i = blockIdx.x * 256 + threadIdx.x; i < NN_ * NN_; i += gridDim.x * 256)
        adjn[i] = (adj[i] - mean) * rstd;
}

// ---------------------------------------------------------------------------
// per-head projections: qh/kh/vh in [h][t][n][hd] slabs for attention locality
// ---------------------------------------------------------------------------
__global__ __launch_bounds__(256) void proj_kernel(const float* __restrict__ val,
                                                   const float* __restrict__ key,
                                                   const float* __restrict__ qry,
                                                   const float* __restrict__ Wv,
                                                   const float* __restrict__ Wk,
                                                   const float* __restrict__ Wq,
                                                   float* __restrict__ vh,
                                                   float* __restrict__ kh,
                                                   float* __restrict__ qh) {
    size_t tid = (size_t)blockIdx.x * 256 + threadIdx.x;   // over NT_*C
    int nt = (int)(tid >> 6);          // row in [N*T)
    int cc = (int)(tid & 63);
    int h = cc >> 4, d = cc & 15;
    int n = nt / TT_, t = nt % TT_;
    const float* xq = qry + (size_t)nt * CC_ + h * HD_;
    const float* xk = key + (size_t)nt * CC_ + h * HD_;
    const float* xv = val + (size_t)nt * CC_ + h * HD_;
    const float* wq = Wq + d * HD_;
    const float* wk = Wk + d * HD_;
    const float* wv = Wv + d * HD_;
    float aq = 0.f, ak = 0.f, av = 0.f;
#pragma unroll
    for (int dd = 0; dd < HD_; ++dd) {
        aq += wq[dd] * xq[dd];
        ak += wk[dd] * xk[dd];
        av += wv[dd] * xv[dd];
    }
    size_t o = ((((size_t)h * TT_ + t) * NN_) + n) * HD_ + d;
    qh[o] = aq; kh[o] = ak; vh[o] = av;
}

// ---------------------------------------------------------------------------
// Flash attention over nodes, per (t,h) slice. grid = T*H*4, block = 8 waves.
// Each wave owns one 16-col q block. We compute the TRANSPOSED score tile
// S^T = K_blk @ Q^T so the softmax axis (k) lands on the register dimension:
// column max / sum = 7 in-lane ops + ONE shfl_xor(16) instead of 32 bpermutes.
// Output accumulates as O^T = V^T @ P^T (d on registers, q on lanes).
// ---------------------------------------------------------------------------
__global__ __launch_bounds__(256) void attn_kernel(const float* __restrict__ qh,
                                                   const float* __restrict__ kh,
                                                   const float* __restrict__ vh,
                                                   float* __restrict__ out) {
    __shared__ float Ks[256 * HD_];
    __shared__ float Vs[256 * HD_];
    __shared__ float Pl[8 * 256];

    const int blk = blockIdx.x;
    const int th = blk >> 2;             // slab index h*T + t
    const int quarter = blk & 3;
    const int h = th / TT_;
    const int t = th % TT_;
    const int tid = threadIdx.x;
    const int wave = tid >> 5;
    const int lane = tid & 31;
    const int lg = lane & 15;            // q column within tile
    const int hi = lane >> 4;            // half-wave select

    const float* Qg = qh + (size_t)th * NN_ * HD_;
    const float* Kg = kh + (size_t)th * NN_ * HD_;
    const float* Vg = vh + (size_t)th * NN_ * HD_;

    const int qbase = (quarter * 8 + wave) * 16;

    // Q^T B-fragments (col = q = lane, vgpr pair covers the d (K-dim) axis)
    v2f bq[4];
#pragma unroll
    for (int ks = 0; ks < 4; ++ks) {
        int d0 = ks * 4 + 2 * hi;
        const float* p = Qg + (size_t)(qbase + lg) * HD_ + d0;
        bq[ks] = v2f{p[0], p[1]};
    }

    float m_run = -3.0e38f, l_run = 0.f;
    v8f o = {};
    float* Pw = Pl + wave * 256;

    for (int p = 0; p < 2; ++p) {
        __syncthreads();
        {   // stage 256 rows of K and V (4096 floats each) into LDS
            const float4* ksrc = (const float4*)(Kg + (size_t)p * 256 * HD_);
            const float4* vsrc = (const float4*)(Vg + (size_t)p * 256 * HD_);
            float4* kd = (float4*)Ks;
            float4* vd = (float4*)Vs;
#pragma unroll
            for (int i = 0; i < 4; ++i) {
                kd[tid + 256 * i] = ksrc[tid + 256 * i];
                vd[tid + 256 * i] = vsrc[tid + 256 * i];
            }
        }
        __syncthreads();

        for (int kb = 0; kb < 16; ++kb) {
            // S^T = K_blk @ Q^T  (A rows = k, cols = q; K-dim = HD -> 4 wmma)
            v8f s = {};
#pragma unroll
            for (int ks = 0; ks < 4; ++ks) {
                int d0 = ks * 4 + 2 * hi;
                const float* kp = Ks + (kb * 16 + lg) * HD_ + d0;
                v2f a = v2f{kp[0], kp[1]};
                s = wmma4(a, bq[ks], s);
            }
            // scaled scores; column (=q) max over the 16 k's of this tile
            float sv[8];
            float cm = -3.0e38f;
#pragma unroll
            for (int j = 0; j < 8; ++j) {
                sv[j] = s[j] * 0.125f;        // 1/sqrt(C)
                cm = fmaxf(cm, sv[j]);
            }
            cm = fmaxf(cm, __shfl_xor(cm, 16));   // combine half-waves
            float mn = fmaxf(m_run, cm);
            float al = __expf(m_run - mn);
            float pj[8];
            float cs = 0.f;
#pragma unroll
            for (int j = 0; j < 8; ++j) {
                pj[j] = __expf(sv[j] - mn);
                cs += pj[j];
            }
            cs += __shfl_xor(cs, 16);
            l_run = l_run * al + cs;
            m_run = mn;
            o *= al;                              // scalar-splat rescale
#pragma unroll
            for (int j = 0; j < 8; ++j)
                Pw[(j + 8 * hi) * 16 + lg] = pj[j];   // P^T row-major (k x q)
            // O^T += V^T @ P^T
#pragma unroll
            for (int ks = 0; ks < 4; ++ks) {
                int kr0 = kb * 16 + ks * 4 + 2 * hi;
                v2f a = v2f{Vs[kr0 * HD_ + lg], Vs[(kr0 + 1) * HD_ + lg]};
                int kk0 = ks * 4 + 2 * hi;
                v2f b = v2f{Pw[kk0 * 16 + lg], Pw[(kk0 + 1) * 16 + lg]};
                o = wmma4(a, b, o);
            }
        }
    }
    // epilogue: O^T(d, q): lane owns q = qbase+lg, d = hi*8 + j (contiguous!)
    const float inv = 1.0f / l_run;
    const int n = qbase + lg;
    float* po = out + ((size_t)n * TT_ + t) * CC_ + h * HD_ + 8 * hi;
    float4 r0 = {o[0] * inv, o[1] * inv, o[2] * inv, o[3] * inv};
    float4 r1 = {o[4] * inv, o[5] * inv, o[6] * inv, o[7] * inv};
    *(float4*)(po)     = r0;
    *(float4*)(po + 4) = r1;
}

// ---------------------------------------------------------------------------
// Generic WMMA GEMM (small K, weight-style B): C[M,Nn] = A[M,K] @ B.T (+bias)
//   BT=1: B is weight [Nn,K] row-major, used transposed (x @ W.T)
//   ACC=1: accumulate into existing C
// one wave per 16x16 output tile
// ---------------------------------------------------------------------------
template <int BT, int RELU, int ACC, int BMOD>
__global__ __launch_bounds__(256) void wgemm_kernel(const float* __restrict__ A,
                                                    const float* __restrict__ B,
                                                    const float* __restrict__ bias,
                                                    float* __restrict__ C,
                                                    int M, int K, int Nn) {
    const int wid = blockIdx.x * (blockDim.x >> 5) + (threadIdx.x >> 5);
    const int tilesN = Nn >> 4;
    if (wid >= (M >> 4) * tilesN) return;          // wave-uniform
    const int mt = wid / tilesN, nt = wid % tilesN;
    const int lane = threadIdx.x & 31;
    const int lg = lane & 15;
    const int hi = lane >> 4;

    v8f c = {};
    if (ACC) {
#pragma unroll
        for (int j = 0; j < 8; ++j)
            c[j] = C[(size_t)(mt * 16 + j + 8 * hi) * Nn + nt * 16 + lg];
    }
    const float* arow = A + (size_t)(mt * 16 + lg) * K;
    for (int k0 = 0; k0 < K; k0 += 4) {
        int kk = k0 + 2 * hi;
        v2f a = v2f{arow[kk], arow[kk + 1]};
        v2f b;
        if (BT) {
            const float* bp = B + (size_t)(nt * 16 + lg) * K + kk;
            b = v2f{bp[0], bp[1]};
        } else {
            int col = nt * 16 + lg;
            b = v2f{B[(size_t)kk * Nn + col], B[(size_t)(kk + 1) * Nn + col]};
        }
        c = wmma4(a, b, c);
    }
#pragma unroll
    for (int j = 0; j < 8; ++j) {
        int row = mt * 16 + j + 8 * hi;
        int col = nt * 16 + lg;
        float v = c[j];
        if (bias) v += bias[BMOD ? (col & 63) : col];
        if (RELU) v = fmaxf(v, 0.f);
        C[(size_t)row * Nn + col] = v;
    }
}

// ---------------------------------------------------------------------------
// adj_n GEMM: C[512,4096] = adjn[512,512] @ X[512,4096] + bias[col&63]
// Block = 8 waves sharing one LDS-staged 16x512 A-slab (lda=516 -> no bank
// conflicts on A-fragment reads); waves cover 8 consecutive n-tiles.
// ---------------------------------------------------------------------------
#define ALDA 516
template <int RELU>
__global__ __launch_bounds__(256) void adjgemm_kernel(const float* __restrict__ A,
                                                      const float* __restrict__ B,
                                                      const float* __restrict__ bias,
                                                      float* __restrict__ C) {
    __shared__ float As[16 * ALDA];
    const int ntGroups = (4096 / 16) / 8;          // 32
    const int mt = blockIdx.x / ntGroups;
    const int ntg = blockIdx.x % ntGroups;
    const int wave = threadIdx.x >> 5;
    const int lane = threadIdx.x & 31;
    const int lg = lane & 15;
    const int hi = lane >> 4;
    const int nt = ntg * 8 + wave;

    // cooperative, coalesced stage of the 16x512 A slab
    for (int i = threadIdx.x; i < 16 * 512; i += 256) {
        int r = i >> 9, k = i & 511;
        As[r * ALDA + k] = A[(size_t)(mt * 16 + r) * 512 + k];
    }
    __syncthreads();

    v8f c = {};
    const int col = nt * 16 + lg;
    for (int k0 = 0; k0 < 512; k0 += 4) {
        int kk = k0 + 2 * hi;
        v2f a = v2f{As[lg * ALDA + kk], As[lg * ALDA + kk + 1]};
        v2f b = v2f{B[(size_t)kk * 4096 + col], B[(size_t)(kk + 1) * 4096 + col]};
        c = wmma4(a, b, c);
    }
#pragma unroll
    for (int j = 0; j < 8; ++j) {
        int row = mt * 16 + j + 8 * hi;
        float v = c[j] + bias[col & 63];
        if (RELU) v = fmaxf(v, 0.f);
        C[(size_t)row * 4096 + col] = v;
    }
}

// ---------------------------------------------------------------------------
// Fused FFN: out = relu(x @ Wf1.T + bf1) @ Wf2.T + bf2 ; 16 rows per block,
// hidden 16x256 kept in LDS.
// ---------------------------------------------------------------------------
__global__ __launch_bounds__(256) void ffn_kernel(const float* __restrict__ x,
                                                  const float* __restrict__ W1,
                                                  const float* __restrict__ b1,
                                                  const float* __restrict__ W2,
                                                  const float* __restrict__ b2,
                                                  float* __restrict__ out) {
    __shared__ float hid[16 * FH_];
    const int rowbase = blockIdx.x * 16;
    const int wave = threadIdx.x >> 5;
    const int lane = threadIdx.x & 31;
    const int lg = lane & 15;
    const int hi = lane >> 4;

    const float* arow = x + (size_t)(rowbase + lg) * CC_;
#pragma unroll
    for (int u = 0; u < 2; ++u) {
        int nt = wave * 2 + u;
        v8f c = {};
#pragma unroll
        for (int k0 = 0; k0 < CC_; k0 += 4) {
            int kk = k0 + 2 * hi;
            v2f a = v2f{arow[kk], arow[kk + 1]};
            const float* bp = W1 + (size_t)(nt * 16 + lg) * CC_ + kk;
            v2f b = v2f{bp[0], bp[1]};
            c = wmma4(a, b, c);
        }
#pragma unroll
        for (int j = 0; j < 8; ++j) {
            int row = j + 8 * hi;
            int col = nt * 16 + lg;
            hid[row * FH_ + col] = fmaxf(c[j] + b1[col], 0.f);
        }
    }
    __syncthreads();
    if (wave < 4) {                                 // wave-uniform branch
        v8f c = {};
        for (int k0 = 0; k0 < FH_; k0 += 4) {
            int kk = k0 + 2 * hi;
            v2f a = v2f{hid[lg * FH_ + kk], hid[lg * FH_ + kk + 1]};
            const float* bp = W2 + (size_t)(wave * 16 + lg) * FH_ + kk;
            v2f b = v2f{bp[0], bp[1]};
            c = wmma4(a, b, c);
        }
#pragma unroll
        for (int j = 0; j < 8; ++j) {
            int row = rowbase + j + 8 * hi;
            int col = wave * 16 + lg;
            out[(size_t)row * CC_ + col] = c[j] + b2[col];
        }
    }
}

// ---------------------------------------------------------------------------
// LayerNorm over last dim (64) of (a + r); one wave per row
// ---------------------------------------------------------------------------
__global__ __launch_bounds__(256) void ln_kernel(const float* __restrict__ a,
                                                 const float* __restrict__ r,
                                                 const float* __restrict__ g,
                                                 const float* __restrict__ b,
                                                 float* __restrict__ out, int rows) {
    int row = blockIdx.x * 8 + (threadIdx.x >> 5);
    if (row >= rows) return;
    int lane = threadIdx.x & 31;
    const float* pa = a + (size_t)row * CC_;
    const float* pr = r + (size_t)row * CC_;
    float v0 = pa[lane] + pr[lane];
    float v1 = pa[lane + 32] + pr[lane + 32];
    float s = v0 + v1;
    for (int off = 16; off > 0; off >>= 1) s += __shfl_xor(s, off);
    float mean = s * (1.f / 64.f);
    float d0 = v0 - mean, d1 = v1 - mean;
    float q = d0 * d0 + d1 * d1;
    for (int off = 16; off > 0; off >>= 1) q += __shfl_xor(q, off);
    float rstd = rsqrtf(q * (1.f / 64.f) + 1e-5f);
    float* po = out + (size_t)row * CC_;
    po[lane]      = d0 * rstd * g[lane]      + b[lane];
    po[lane + 32] = d1 * rstd * g[lane + 32] + b[lane + 32];
}

// ---------------------------------------------------------------------------
// gated fusion: out = sigmoid(z)*o1 + (1-sigmoid(z))*o2
// ---------------------------------------------------------------------------
__global__ __launch_bounds__(256) void blend_kernel(const float* __restrict__ z,
                                                    const float* __restrict__ o1,
                                                    const float* __restrict__ o2,
                                                    float* __restrict__ out, int n) {
    for (int i = blockIdx.x * 256 + threadIdx.x; i < n; i += gridDim.x * 256) {
        float gate = 1.f / (1.f + __expf(-z[i]));
        out[i] = gate * o1[i] + (1.f - gate) * o2[i];
    }
}

// ---------------------------------------------------------------------------
extern "C" void kernel_launch(void* const* d_in, const int* in_sizes, int n_in,
                              void* d_out, int out_size, void* d_ws, size_t ws_size,
                              hipStream_t stream) {
    const float* value = (const float*)d_in[0];
    const float* key_t = (const float*)d_in[1];
    const float* query = (const float*)d_in[2];
    const float* adj   = (const float*)d_in[3];
    const float* Wv = (const float*)d_in[4];
    const float* Wk = (const float*)d_in[5];
    const float* Wq = (const float*)d_in[6];
    const float* Wo = (const float*)d_in[7];
    const float* bo = (const float*)d_in[8];
    const float* g1 = (const float*)d_in[9];
    const float* b1 = (const float*)d_in[10];
    const float* g2 = (const float*)d_in[11];
    const float* b2 = (const float*)d_in[12];
    const float* Wf1 = (const float*)d_in[13];
    const float* bf1 = (const float*)d_in[14];
    const float* Wf2 = (const float*)d_in[15];
    const float* bf2 = (const float*)d_in[16];
    const float* Wg1 = (const float*)d_in[17];
    const float* bg1 = (const float*)d_in[18];
    const float* Wg2 = (const float*)d_in[19];
    const float* bg2 = (const float*)d_in[20];
    const float* Wo1 = (const float*)d_in[21];
    const float* bo1 = (const float*)d_in[22];
    const float* Wo2 = (const float*)d_in[23];
    const float* bo2 = (const float*)d_in[24];
    float* out = (float*)d_out;

    float* ws = (float*)d_ws;
    float* S0 = ws;               // qh      -> g1x -> z
    float* S1 = ws + 1 * SLOT;    // kh      -> gcn hidden
    float* S2 = ws + 2 * SLOT;    // vh      -> g2x
    float* S3 = ws + 3 * SLOT;    // attnraw -> ffn out -> out2
    float* S4 = ws + 4 * SLOT;    // y       -> out1
    float* S5 = ws + 5 * SLOT;    // x (post-LN1)
    float* ADJN  = ws + 6 * SLOT;
    float* STATS = ADJN + (size_t)NN_ * NN_;

    // 0. adj instance norm
    adj_stats_kernel<<<1, 256, 0, stream>>>(adj, STATS);
    adj_norm_kernel<<<256, 256, 0, stream>>>(adj, STATS, ADJN);

    // 1. per-head projections
    proj_kernel<<<(NT_ * CC_) / 256, 256, 0, stream>>>(value, key_t, query,
                                                       Wv, Wk, Wq, S2, S1, S0);

    // 2. spatial flash attention per (t,h)
    attn_kernel<<<TT_ * HH_ * 4, 256, 0, stream>>>(S0, S1, S2, S3);

    // 3. output projection: y = attn @ Wo.T + bo
    wgemm_kernel<1, 0, 0, 0><<<(NT_ / 16) * (CC_ / 16) / 8, 256, 0, stream>>>(
        S3, Wo, bo, S4, NT_, CC_, CC_);

    // 4. x = LN1(y + query)
    ln_kernel<<<NT_ / 8, 256, 0, stream>>>(S4, query, g1, b1, S5, NT_);

    // 5. fused FFN -> S3
    ffn_kernel<<<NT_ / 16, 256, 0, stream>>>(S5, Wf1, bf1, Wf2, bf2, S3);

    // 6. out1 = LN2(ffn + x) -> S4
    ln_kernel<<<NT_ / 8, 256, 0, stream>>>(S3, S5, g2, b2, S4, NT_);

    // 7. GCN layer 1: g1x = query @ Wg1.T -> S0
    wgemm_kernel<1, 0, 0, 0><<<(NT_ / 16) * (CC_ / 16) / 8, 256, 0, stream>>>(
        query, Wg1, nullptr, S0, NT_, CC_, CC_);
    //    h = relu(adjn @ g1x + bg1) -> S1   (adjn:512x512, X:512x4096)
    adjgemm_kernel<1><<<(NN_ / 16) * ((TT_ * CC_ / 16) / 8), 256, 0, stream>>>(
        ADJN, S0, bg1, S1);
    // 8. GCN layer 2: g2x = h @ Wg2.T -> S2
    wgemm_kernel<1, 0, 0, 0><<<(NT_ / 16) * (CC_ / 16) / 8, 256, 0, stream>>>(
        S1, Wg2, nullptr, S2, NT_, CC_, CC_);
    //    out2 = adjn @ g2x + bg2 -> S3
    adjgemm_kernel<0><<<(NN_ / 16) * ((TT_ * CC_ / 16) / 8), 256, 0, stream>>>(
        ADJN, S2, bg2, S3);

    // 9. gate pre-activation: z = out1 @ Wo1.T + bo1 ; z += out2 @ Wo2.T + bo2
    wgemm_kernel<1, 0, 0, 0><<<(NT_ / 16) * (CC_ / 16) / 8, 256, 0, stream>>>(
        S4, Wo1, bo1, S0, NT_, CC_, CC_);
    wgemm_kernel<1, 0, 1, 0><<<(NT_ / 16) * (CC_ / 16) / 8, 256, 0, stream>>>(
        S3, Wo2, bo2, S0, NT_, CC_, CC_);

    // 10. gated fusion
    blend_kernel<<<2048, 256, 0, stream>>>(S0, S4, S3, out, NT_ * CC_);
}